// LocalCrossCorrelation3D_87582973100653
// MI455X (gfx1250) — compile-verified
//
#include <hip/hip_runtime.h>

typedef float v2f __attribute__((ext_vector_type(2)));
typedef float v8f __attribute__((ext_vector_type(8)));

#define D_IN   96
#define H_IN   192
#define W_IN   192
#define D_OUT  81
#define WIN_D  16
#define WIN_SZ 1296.0f
#define EPS_F  3.0590232050182579e-7f   // e^-15
#define TILE   16
#define CH_D   27                        // output depths per chunk (81 = 3*27)
#define SLICES (CH_D + WIN_D - 1)        // 42 input slices per chunk

// -------------------------------------------------------------------------
// Fused LNCC: products + separable (9W x 9H x 16D) box sums + cc, one pass.
// H-direction 9-tap done as S = A_band(16x24) * T(24x16) on the matrix pipe
// via V_WMMA_F32_16X16X4_F32 (6 chunks of K=4), co-executing with VALU work.
// -------------------------------------------------------------------------
__global__ __launch_bounds__(32)
void lncc_fused(const float* __restrict__ I, const float* __restrict__ J,
                float* __restrict__ cc_out, float* __restrict__ part) {
  // LDS: 16-slice ring of per-lane window sums (5ch x 256 voxels) + T staging
  __shared__ float ring[16 * 5 * 256];      // 80 KB
  __shared__ float tbuf[5 * 24 * 16];       // 7.5 KB

  const int lane = threadIdx.x;
  const int col  = lane & 15;               // N for B/C/D, M for A operand
  const int half = lane >> 4;

  int task = blockIdx.x;
  const int tw = task % 12; task /= 12;
  const int th = task % 12; task /= 12;
  const int dc = task % 3;  const int b = task / 3;
  const int w0 = tw * TILE, h0 = th * TILE;
  const int dstart = dc * CH_D;

  // Constant band matrix A (16x24): A[m][k] = 1 iff m <= k <= m+8.
  // f32 16x4 A layout: lanes 0-15 hold K = k0+0 (v.x), k0+1 (v.y);
  //                    lanes 16-31 hold K = k0+2 (v.x), k0+3 (v.y).
  v2f Ac[6];
#pragma unroll
  for (int kk = 0; kk < 6; ++kk) {
    const int ka = 4 * kk + (half ? 2 : 0);
    Ac[kk].x = (ka     >= col && ka     <= col + 8) ? 1.0f : 0.0f;
    Ac[kk].y = (ka + 1 >= col && ka + 1 <= col + 8) ? 1.0f : 0.0f;
  }

  float run[5][8];
#pragma unroll
  for (int c = 0; c < 5; ++c)
#pragma unroll
    for (int i = 0; i < 8; ++i) run[c][i] = 0.0f;

  float lsum = 0.0f;

  for (int d_idx = 0; d_idx < SLICES; ++d_idx) {
    const int d = dstart + d_idx;

    // ---- Phase A: one halo row per lane -> products -> sliding 9-wide W sums
    if (lane < 24) {
      const int  r   = lane;
      const int  h   = h0 - 4 + r;
      const bool inH = (h >= 0) && (h < H_IN);
      const size_t rowBase =
          ((size_t)((b * D_IN + d) * H_IN + (inH ? h : 0))) * W_IN;

      float Iv[24], Jv[24];
#pragma unroll
      for (int j = 0; j < 6; ++j) {
        const int iw0 = w0 - 4 + 4 * j;     // 4-aligned; OOB is all-or-nothing
        float4 ai = make_float4(0.f, 0.f, 0.f, 0.f);
        float4 aj = make_float4(0.f, 0.f, 0.f, 0.f);
        if (inH && iw0 >= 0 && iw0 + 4 <= W_IN) {
          ai = *(const float4*)(I + rowBase + iw0);
          aj = *(const float4*)(J + rowBase + iw0);
        }
        Iv[4*j+0] = ai.x; Iv[4*j+1] = ai.y; Iv[4*j+2] = ai.z; Iv[4*j+3] = ai.w;
        Jv[4*j+0] = aj.x; Jv[4*j+1] = aj.y; Jv[4*j+2] = aj.z; Jv[4*j+3] = aj.w;
      }
      if (inH && d_idx + 1 < SLICES) {      // pull next depth slice toward L2
        __builtin_prefetch(I + rowBase + (size_t)H_IN * W_IN + w0, 0, 0);
        __builtin_prefetch(J + rowBase + (size_t)H_IN * W_IN + w0, 0, 0);
      }

#pragma unroll
      for (int c = 0; c < 5; ++c) {
        float p[24];
#pragma unroll
        for (int q = 0; q < 24; ++q) {
          const float pi = Iv[q], pj = Jv[q];
          p[q] = (c == 0) ? pi
               : (c == 1) ? pj
               : (c == 2) ? pi * pi
               : (c == 3) ? pj * pj
                          : pi * pj;
        }
        float tq[16];
        float t = 0.0f;
#pragma unroll
        for (int q = 0; q < 9; ++q) t += p[q];
        tq[0] = t;
#pragma unroll
        for (int w = 1; w < 16; ++w) { t += p[w + 8] - p[w - 1]; tq[w] = t; }

        float4* dstp = (float4*)&tbuf[(c * 24 + r) * 16];
        dstp[0] = make_float4(tq[0],  tq[1],  tq[2],  tq[3]);
        dstp[1] = make_float4(tq[4],  tq[5],  tq[6],  tq[7]);
        dstp[2] = make_float4(tq[8],  tq[9],  tq[10], tq[11]);
        dstp[3] = make_float4(tq[12], tq[13], tq[14], tq[15]);
      }
    }
    __syncthreads();

    // ---- Phase B: H-direction 9-tap as WMMA; D-direction sliding window
    const int slot = d_idx & 15;
#pragma unroll
    for (int c = 0; c < 5; ++c) {
      v8f s = {0.f, 0.f, 0.f, 0.f, 0.f, 0.f, 0.f, 0.f};
#pragma unroll
      for (int kk = 0; kk < 6; ++kk) {
        const int kb = 4 * kk + (half ? 2 : 0);   // B 4x16 f32 layout (mirror of A)
        v2f bb;
        bb.x = tbuf[(c * 24 + kb    ) * 16 + col];
        bb.y = tbuf[(c * 24 + kb + 1) * 16 + col];
        s = __builtin_amdgcn_wmma_f32_16x16x4_f32(false, Ac[kk], false, bb,
                                                  (short)0, s, false, false);
      }
      // ring: run += s_new - s(d-16); per-lane-private LDS slots
      float4* rp = (float4*)&ring[(slot * 5 + c) * 256 + lane * 8];
      if (d_idx >= WIN_D) {
        const float4 o0 = rp[0], o1 = rp[1];
        run[c][0] -= o0.x; run[c][1] -= o0.y; run[c][2] -= o0.z; run[c][3] -= o0.w;
        run[c][4] -= o1.x; run[c][5] -= o1.y; run[c][6] -= o1.z; run[c][7] -= o1.w;
      }
#pragma unroll
      for (int i = 0; i < 8; ++i) run[c][i] += s[i];
      rp[0] = make_float4(s[0], s[1], s[2], s[3]);
      rp[1] = make_float4(s[4], s[5], s[6], s[7]);
    }

    // ---- emit cc once a full 16-deep window is resident
    if (d_idx >= WIN_D - 1) {
      const int dout = dstart + d_idx - (WIN_D - 1);
#pragma unroll
      for (int i = 0; i < 8; ++i) {
        const float Is  = run[0][i], Js  = run[1][i];
        const float I2s = run[2][i], J2s = run[3][i], IJs = run[4][i];
        const float uI = Is * (1.0f / WIN_SZ), uJ = Js * (1.0f / WIN_SZ);
        float cross = IJs - uJ * Is - uI * Js + uI * uJ * WIN_SZ;
        float Ivar  = I2s - 2.0f * uI * Is + uI * uI * WIN_SZ;
        float Jvar  = J2s - 2.0f * uJ * Js + uJ * uJ * WIN_SZ;
        if (!((Ivar * Jvar) > EPS_F)) { cross = 1.0f; Ivar = 1.0f; Jvar = 1.0f; }
        const float cc = cross * cross / (Ivar * Jvar + EPS_F);
        const int m = i + (half ? 8 : 0);    // C/D layout: VGPR i -> rows i, i+8
        cc_out[(((size_t)b * D_OUT + dout) * H_IN + (h0 + m)) * W_IN + (w0 + col)] = cc;
        lsum += cc;
      }
    }
    __syncthreads();   // protect tbuf WAR for next slice
  }

  atomicAdd(&part[b], lsum);
}

__global__ void lncc_zero(float* ws) {
  if (threadIdx.x < 2) ws[threadIdx.x] = 0.0f;
}

__global__ void lncc_finalize(const float* __restrict__ ws, float* __restrict__ out) {
  if (threadIdx.x < 2)
    out[threadIdx.x] =
        1.0f - ws[threadIdx.x] * (1.0f / ((float)D_OUT * H_IN * W_IN));
}

extern "C" void kernel_launch(void* const* d_in, const int* in_sizes, int n_in,
                              void* d_out, int out_size, void* d_ws, size_t ws_size,
                              hipStream_t stream) {
  (void)in_sizes; (void)n_in; (void)out_size; (void)ws_size;
  const float* I = (const float*)d_in[0];
  const float* J = (const float*)d_in[1];
  float* out = (float*)d_out;          // [0..1] loss, [2..] cc flat
  float* ws  = (float*)d_ws;

  lncc_zero<<<1, 32, 0, stream>>>(ws);
  // tasks: 2 batch x 3 depth-chunks x 12 x 12 tiles = 864 single-wave blocks
  lncc_fused<<<dim3(864), dim3(32), 0, stream>>>(I, J, out + 2, ws);
  lncc_finalize<<<1, 32, 0, stream>>>(ws, out);
}